// GCN_28243704939204
// MI455X (gfx1250) — compile-verified
//
#include <hip/hip_runtime.h>
#include <hip/hip_bf16.h>

// ---------------------------------------------------------------------------
// GCN 2-layer forward for MI455X (gfx1250, wave32).
//
// Pipeline:
//   1) h1   = X @ W1 + b1                      (WMMA f32 16x16x4)
//   2) agg1 = segment_sum(h1[src] -> dst), deg = segment_sum(1 -> dst)
//   3) h1n  = relu(agg1 / max(deg,1))          (in-place over h1 buffer)
//   4) h2   = h1n @ W2 + b2                    (WMMA f32 16x16x4, into agg1 buf)
//   5) out  = segment_sum(h2[src] -> dst) / max(deg,1)   (deg reused)
//
// N_NODES=50000 = 3125*16 exactly -> no M predication, EXEC all-1s for WMMA.
// ---------------------------------------------------------------------------

typedef __attribute__((ext_vector_type(2))) float v2f;
typedef __attribute__((ext_vector_type(8))) float v8f;

#define GCN_NODES 50000
#define GCN_DIN   128
#define GCN_DHID  256
#define GCN_DOUT  128

// ---------------------------------------------------------------------------
// GEMM + bias: H[M,N] = X[M,K] @ W[K,N] + b[N]
// One wave (32 threads) computes a 16(M) x 64(N) tile: 4 accumulators of
// v_wmma_f32_16x16x4_f32, K stepped by 4. A-fragment reused across the 4
// N-subtiles.
//
// ISA layouts (CDNA5 §7.12.2, 32-bit):
//   A 16x4 : lane l<16 -> {A[m][k], A[m][k+1]} ; lane l+16 -> {A[m][k+2], A[m][k+3]}
//   B 4x16 : lane n<16 -> v0=B[k][n],  v1=B[k+1][n];
//            lane n+16 -> v0=B[k+2][n],v1=B[k+3][n]
//   C/D    : vgpr i, lanes 0-15 -> row i, col=lane; lanes 16-31 -> row i+8
// ---------------------------------------------------------------------------
__global__ void gcn_gemm_bias_wmma(const float* __restrict__ X,
                                   const float* __restrict__ W,
                                   const float* __restrict__ bias,
                                   float* __restrict__ H,
                                   int K, int N) {
    const int lane = threadIdx.x;      // 0..31 (wave32)
    const int half = lane >> 4;        // 0 or 1
    const int l    = lane & 15;

    const int mBase = blockIdx.x * 16;
    const int nBase = blockIdx.y * 64;

    const float* arow = X + (size_t)(mBase + l) * (size_t)K;

    // Accumulators initialized with the bias (bias depends only on column,
    // and in the C/D layout every VGPR of lane `lane` maps to column
    // nBase + t*16 + (lane&15)).
    v8f acc[4];
#pragma unroll
    for (int t = 0; t < 4; ++t) {
        const float bv = bias[nBase + t * 16 + l];
#pragma unroll
        for (int i = 0; i < 8; ++i) acc[t][i] = bv;
    }

    for (int k = 0; k < K; k += 4) {
        // A fragment: aligned float2 (k is a multiple of 4, half*2 keeps it even)
        const v2f a = *(const v2f*)(arow + k + 2 * half);

        const int krow = k + 2 * half;
#pragma unroll
        for (int t = 0; t < 4; ++t) {
            const int col = nBase + t * 16 + l;
            v2f b;
            b.x = W[(size_t)(krow)     * (size_t)N + col];
            b.y = W[(size_t)(krow + 1) * (size_t)N + col];
            acc[t] = __builtin_amdgcn_wmma_f32_16x16x4_f32(
                /*neg_a=*/false, a, /*neg_b=*/false, b,
                /*c_mod=*/(short)0, acc[t],
                /*reuse_a=*/false, /*reuse_b=*/false);
        }
    }

    // Store D tiles.
#pragma unroll
    for (int t = 0; t < 4; ++t) {
        const int col = nBase + t * 16 + l;
#pragma unroll
        for (int i = 0; i < 8; ++i) {
            const int row = mBase + i + 8 * half;
            H[(size_t)row * (size_t)N + col] = acc[t][i];
        }
    }
}

// ---------------------------------------------------------------------------
// Edge-parallel scatter-add: agg[dst] += h[src] (C columns, C % 4 == 0).
// One thread per (edge, 4-column chunk); consecutive threads take consecutive
// column chunks of the same edge -> coalesced float4 gather, then four
// global_atomic_add_f32 (no-return, STOREcnt). Degree accumulated once/edge.
// ---------------------------------------------------------------------------
__global__ void gcn_scatter_add(const float* __restrict__ h,
                                const int* __restrict__ edges,
                                float* __restrict__ agg,
                                float* __restrict__ deg,
                                long long nEdges, int C, int addDeg) {
    const int chunks = C >> 2;
    const long long tid = (long long)blockIdx.x * blockDim.x + threadIdx.x;
    const long long total = nEdges * (long long)chunks;
    if (tid >= total) return;

    const long long e  = tid / chunks;
    const int       c4 = (int)(tid - e * (long long)chunks) << 2;

    const int src = edges[2 * e];
    const int dst = edges[2 * e + 1];

    const float4 v = *(const float4*)(h + (size_t)src * (size_t)C + c4);
    float* p = agg + (size_t)dst * (size_t)C + c4;
    atomicAdd(p + 0, v.x);
    atomicAdd(p + 1, v.y);
    atomicAdd(p + 2, v.z);
    atomicAdd(p + 3, v.w);

    if (addDeg && c4 == 0) atomicAdd(deg + dst, 1.0f);
}

// out[i] = agg[i] / max(deg[node],1), optional relu. Safe in-place.
__global__ void gcn_normalize(const float* __restrict__ agg,
                              const float* __restrict__ deg,
                              float* __restrict__ out,
                              long long n, int C, int doRelu) {
    const long long i = (long long)blockIdx.x * blockDim.x + threadIdx.x;
    if (i >= n) return;
    const long long node = i / C;
    const float d = fmaxf(deg[node], 1.0f);
    float v = agg[i] / d;
    if (doRelu) v = fmaxf(v, 0.0f);
    out[i] = v;
}

__global__ void gcn_zero_f32(float* __restrict__ p, long long n) {
    const long long i = (long long)blockIdx.x * blockDim.x + threadIdx.x;
    if (i < n) p[i] = 0.0f;
}

// ---------------------------------------------------------------------------
extern "C" void kernel_launch(void* const* d_in, const int* in_sizes, int n_in,
                              void* d_out, int out_size, void* d_ws, size_t ws_size,
                              hipStream_t stream) {
    const float* X  = (const float*)d_in[0];   // [50000,128]
    const int*   E  = (const int*)  d_in[1];   // [800000,2] (src,dst)
    const float* W1 = (const float*)d_in[2];   // [128,256]
    const float* b1 = (const float*)d_in[3];   // [256]
    const float* W2 = (const float*)d_in[4];   // [256,128]
    const float* b2 = (const float*)d_in[5];   // [128]
    float* out = (float*)d_out;                // [50000,128]

    const long long nEdges = (long long)in_sizes[1] / 2;

    // Workspace carve-up.
    char* ws = (char*)d_ws;
    float* h1   = (float*)ws;                                   // 50000*256 f32
    float* agg1 = (float*)(ws + (size_t)GCN_NODES * GCN_DHID * 4);   // 50000*256 f32
    float* deg  = (float*)(ws + (size_t)GCN_NODES * GCN_DHID * 8);   // 50000 f32
    float* h2   = agg1;  // reuse agg1 buffer for layer-2 GEMM output

    const long long nH1  = (long long)GCN_NODES * GCN_DHID;     // 12.8M
    const long long nOut = (long long)GCN_NODES * GCN_DOUT;     // 6.4M

    // 1) h1 = X @ W1 + b1    (M=50000 -> 3125 tiles, N=256 -> 4 x 64-wide strips)
    gcn_gemm_bias_wmma<<<dim3(GCN_NODES / 16, GCN_DHID / 64), 32, 0, stream>>>(
        X, W1, b1, h1, GCN_DIN, GCN_DHID);

    // 2) zero agg1 + deg (contiguous), then scatter h1 by edges (+deg).
    {
        const long long nz = nH1 + GCN_NODES;
        gcn_zero_f32<<<(int)((nz + 255) / 256), 256, 0, stream>>>(agg1, nz);
    }
    {
        const long long tot = nEdges * (GCN_DHID / 4);
        gcn_scatter_add<<<(int)((tot + 255) / 256), 256, 0, stream>>>(
            h1, E, agg1, deg, nEdges, GCN_DHID, /*addDeg=*/1);
    }

    // 3) h1n = relu(agg1 / max(deg,1))  -> overwrite h1 buffer
    gcn_normalize<<<(int)((nH1 + 255) / 256), 256, 0, stream>>>(
        agg1, deg, h1, nH1, GCN_DHID, /*doRelu=*/1);

    // 4) h2 = h1n @ W2 + b2  (N=128 -> 2 x 64-wide strips), into agg1 buffer
    gcn_gemm_bias_wmma<<<dim3(GCN_NODES / 16, GCN_DOUT / 64), 32, 0, stream>>>(
        h1, W2, b2, h2, GCN_DHID, GCN_DOUT);

    // 5) zero d_out, scatter h2 by edges (deg reused), normalize in place.
    gcn_zero_f32<<<(int)((nOut + 255) / 256), 256, 0, stream>>>(out, nOut);
    {
        const long long tot = nEdges * (GCN_DOUT / 4);
        gcn_scatter_add<<<(int)((tot + 255) / 256), 256, 0, stream>>>(
            h2, E, out, deg, nEdges, GCN_DOUT, /*addDeg=*/0);
    }
    gcn_normalize<<<(int)((nOut + 255) / 256), 256, 0, stream>>>(
        out, deg, out, nOut, GCN_DOUT, /*doRelu=*/0);
}